// MultiHeadAttention_49151605735523
// MI455X (gfx1250) — compile-verified
//
#include <hip/hip_runtime.h>
#include <stdint.h>

typedef float v2f __attribute__((ext_vector_type(2)));
typedef float v8f __attribute__((ext_vector_type(8)));

static constexpr int B_ = 4;
static constexpr int T_ = 2048;
static constexpr int D_ = 1024;
static constexpr int H_ = 16;
static constexpr int DK_ = 64;

#define KSTR 36   // LDS row stride (dwords) for 32-wide K-chunk tiles (+4 pad)
#define LSTR 68   // LDS row stride (dwords) for 64-wide tiles (+4 pad)
#define PSTR 36   // stride for per-wave P staging (16 x 32 halves)

__device__ __forceinline__ v8f wmma_f32(v2f a, v2f b, v8f c) {
  // V_WMMA_F32_16X16X4_F32 : D = A(16x4) * B(4x16) + C(16x16), fp32
  return __builtin_amdgcn_wmma_f32_16x16x4_f32(
      false, a, false, b, (short)0, c, false, false);
}

// Low 32 bits of a flat LDS pointer == LDS byte offset (shared aperture).
__device__ __forceinline__ uint32_t lds_off(const void* p) {
  return (uint32_t)(uintptr_t)p;
}

// CDNA5 async global->LDS copy (ASYNCcnt). Per-lane: copies 16 bytes from the global address in %1 to the LDS address in %0.
__device__ __forceinline__ void async_ld_b128(uint32_t ldsAddr, const float* g) {
  asm volatile("global_load_async_to_lds_b128 %0, %1, off"
               :: "v"(ldsAddr), "v"((uint64_t)(uintptr_t)g)
               : "memory");
}
__device__ __forceinline__ void wait_async0() {
  asm volatile("s_wait_asynccnt 0x0" ::: "memory");
}

// ---------------------------------------------------------------------------
// Generic GEMM:  Out[z] = A[z] (MxK, rm) * W[z]^T (W is NxK, rm) + bias, *scale
// Block: 256 threads / 8 waves; 128x64 output tile; K streamed in 32-chunks,
// double-buffered through LDS with async global->LDS loads. Each wave owns a
// 2x2 grid of 16x16 WMMA tiles (4 WMMA per 4 b64 frag loads).
// ---------------------------------------------------------------------------
__global__ __launch_bounds__(256)
void gemm_xwT_kernel(const float* __restrict__ A, const float* __restrict__ W,
                     const float* __restrict__ bias, float* __restrict__ Out,
                     int M, int N, int K, float scale,
                     int zDivA, long long aStride, int zModW, long long wStride,
                     long long bStride, long long oStride)
{
  __shared__ __align__(16) float Xs[2][128 * KSTR];
  __shared__ __align__(16) float Ws[2][64 * KSTR];

  const int z = blockIdx.z;
  const float* Ab = A + (long long)(z / zDivA) * aStride;
  const float* Wb = W + (long long)(z % zModW) * wStride;
  const float* Bb = bias + (long long)(z % zModW) * bStride;
  float* Ob = Out + (long long)z * oStride;

  const int m0 = blockIdx.x * 128;
  const int n0 = blockIdx.y * 64;

  const int tid  = threadIdx.x;
  const int lane = tid & 31;
  const int wave = tid >> 5;
  const int half = lane >> 4;         // selects K pair within WMMA frag
  const int l16  = lane & 15;

  const int mi0 = (wave & 3) * 2;     // row tiles mi0, mi0+1 (of 8)
  const int ni0 = (wave >> 2) * 2;    // col tiles ni0, ni0+1 (of 4)

  // async loader mapping: 32 rows x 8 float4-cols per pass
  const int lr  = tid >> 3;           // 0..31
  const int lc4 = (tid & 7) * 4;      // 0..28

  const uint32_t xsB[2] = { lds_off(&Xs[0][0]), lds_off(&Xs[1][0]) };
  const uint32_t wsB[2] = { lds_off(&Ws[0][0]), lds_off(&Ws[1][0]) };

  auto issue = [&](int kc, int buf) {
#pragma unroll
    for (int i = 0; i < 4; ++i) {
      const int r = lr + i * 32;
      async_ld_b128(xsB[buf] + (uint32_t)(r * KSTR + lc4) * 4,
                    Ab + (long long)(m0 + r) * K + kc + lc4);
    }
#pragma unroll
    for (int i = 0; i < 2; ++i) {
      const int r = lr + i * 32;
      async_ld_b128(wsB[buf] + (uint32_t)(r * KSTR + lc4) * 4,
                    Wb + (long long)(n0 + r) * K + kc + lc4);
    }
  };

  v8f acc[2][2] = {};

  issue(0, 0);
  const int nchunks = K / 32;
  for (int c = 0; c < nchunks; ++c) {
    wait_async0();        // own async writes landed
    __syncthreads();      // (lowered with s_wait_dscnt 0) -> buffer published,
                          // and previous buffer's readers are drained
    if (c + 1 < nchunks) issue((c + 1) * 32, (c + 1) & 1);

    const float* Xc = Xs[c & 1];
    const float* Wc = Ws[c & 1];
    const int am0 = (mi0 * 16 + l16) * KSTR;
    const int am1 = am0 + 16 * KSTR;
    const int bn0 = (ni0 * 16 + l16) * KSTR;
    const int bn1 = bn0 + 16 * KSTR;
#pragma unroll
    for (int kb = 0; kb < 32; kb += 4) {
      const int kk = kb + half * 2;
      const v2f a0 = *(const v2f*)(Xc + am0 + kk);
      const v2f a1 = *(const v2f*)(Xc + am1 + kk);
      const v2f b0 = *(const v2f*)(Wc + bn0 + kk);
      const v2f b1 = *(const v2f*)(Wc + bn1 + kk);
      acc[0][0] = wmma_f32(a0, b0, acc[0][0]);
      acc[0][1] = wmma_f32(a0, b1, acc[0][1]);
      acc[1][0] = wmma_f32(a1, b0, acc[1][0]);
      acc[1][1] = wmma_f32(a1, b1, acc[1][1]);
    }
  }

  // Epilogue: C/D layout -> lane l16 = column, vgpr r = row (+8 for high half)
#pragma unroll
  for (int t = 0; t < 2; ++t) {
    const int row0 = m0 + (mi0 + t) * 16 + half * 8;
#pragma unroll
    for (int u = 0; u < 2; ++u) {
      const int col = n0 + (ni0 + u) * 16 + l16;
      const float bv = Bb[col];
#pragma unroll
      for (int r = 0; r < 8; ++r)
        Ob[(long long)(row0 + r) * N + col] = (acc[t][u][r] + bv) * scale;
    }
  }
}

// ---------------------------------------------------------------------------
// Flash attention (causal, online softmax). Q pre-scaled by 1/sqrt(DK).
// Block: 256 threads / 8 waves; each wave owns 16 q-rows (128 per block).
// K/V tiles staged via async global->LDS; P transposed through wave-private
// LDS (C/D layout -> A-frag layout) for the P*V WMMA.
// ---------------------------------------------------------------------------
__global__ __launch_bounds__(256)
void flash_attn_kernel(const float* __restrict__ Q, const float* __restrict__ Kg,
                       const float* __restrict__ V, float* __restrict__ AttOut)
{
  __shared__ __align__(16) float Ks[64 * LSTR];
  __shared__ __align__(16) float Vs[64 * LSTR];
  __shared__ __align__(16) float Ps[8 * 16 * PSTR];

  const int bh = blockIdx.y;
  const int b  = bh >> 4;
  const int h  = bh & 15;
  const int q0 = blockIdx.x * 128;

  const int tid  = threadIdx.x;
  const int lane = tid & 31;
  const int wave = tid >> 5;
  const int half = lane >> 4;
  const int l16  = lane & 15;

  const float* Qb = Q  + (long long)bh * T_ * DK_;
  const float* Kb = Kg + (long long)bh * T_ * DK_;
  const float* Vb = V  + (long long)bh * T_ * DK_;

  const uint32_t ksB = lds_off(&Ks[0]);
  const uint32_t vsB = lds_off(&Vs[0]);

  // Q A-frags in registers: 16 K-steps of 4 (documented 16x4 f32 A layout)
  v2f qf[16];
  {
    const float* qrow = Qb + (long long)(q0 + wave * 16 + l16) * DK_;
#pragma unroll
    for (int kb = 0; kb < 16; ++kb)
      qf[kb] = *(const v2f*)(qrow + kb * 4 + half * 2);
  }

  v8f o0 = {}, o1 = {}, o2 = {}, o3 = {};
  float mrow[8], lsum[8];
#pragma unroll
  for (int r = 0; r < 8; ++r) { mrow[r] = -__builtin_inff(); lsum[r] = 0.f; }

  float* Pw = &Ps[wave * 16 * PSTR];
  const int jblocks = q0 / 64 + 2;   // key blocks with jbase <= q0+127

  for (int jb = 0; jb < jblocks; ++jb) {
    const int jbase = jb * 64;
    __syncthreads();   // previous tile's readers drained (dscnt flushed)
    {
      const int lrow  = tid >> 4;
      const int lcol4 = (tid & 15) * 4;
#pragma unroll
      for (int i = 0; i < 4; ++i) {
        const int r = lrow + i * 16;
        async_ld_b128(ksB + (uint32_t)(r * LSTR + lcol4) * 4,
                      Kb + (long long)(jbase + r) * DK_ + lcol4);
        async_ld_b128(vsB + (uint32_t)(r * LSTR + lcol4) * 4,
                      Vb + (long long)(jbase + r) * DK_ + lcol4);
      }
    }
    wait_async0();
    __syncthreads();

    // S = Q * K^T : 4 tiles of 16x16 per wave
    v8f s[4];
#pragma unroll
    for (int ni = 0; ni < 4; ++ni) {
      v8f acc = {};
      const int n = ni * 16 + l16;
#pragma unroll
      for (int kb = 0; kb < 16; ++kb) {
        const v2f bf = *(const v2f*)(&Ks[n * LSTR + kb * 4 + half * 2]);
        acc = wmma_f32(qf[kb], bf, acc);
      }
      s[ni] = acc;
    }

    // Causal mask + online softmax (row stats replicated over 16-lane group)
    const int rowbase = q0 + wave * 16 + half * 8;
    float mnew[8];
#pragma unroll
    for (int r = 0; r < 8; ++r) {
      const int i = rowbase + r;
      float mx = -__builtin_inff();
#pragma unroll
      for (int ni = 0; ni < 4; ++ni) {
        const int j = jbase + ni * 16 + l16;
        if (j > i) s[ni][r] = -__builtin_inff();
        mx = fmaxf(mx, s[ni][r]);
      }
#pragma unroll
      for (int off = 1; off < 16; off <<= 1)
        mx = fmaxf(mx, __shfl_xor(mx, off, 16));
      mnew[r] = fmaxf(mrow[r], mx);
      const float alpha = __expf(mrow[r] - mnew[r]);
      mrow[r] = mnew[r];
      lsum[r] *= alpha;
      o0[r] *= alpha; o1[r] *= alpha; o2[r] *= alpha; o3[r] *= alpha;
    }
#pragma unroll
    for (int r = 0; r < 8; ++r) {
      float rs = 0.f;
#pragma unroll
      for (int ni = 0; ni < 4; ++ni) {
        const float p = __expf(s[ni][r] - mnew[r]);
        s[ni][r] = p;
        rs += p;
      }
#pragma unroll
      for (int off = 1; off < 16; off <<= 1)
        rs += __shfl_xor(rs, off, 16);
      lsum[r] += rs;
    }

    // O += P * V in two key-halves through wave-private LDS.
    // DS ops are in-order within a wave: no barrier needed for Pw.
#pragma unroll
    for (int hv = 0; hv < 2; ++hv) {
#pragma unroll
      for (int r = 0; r < 8; ++r) {
        Pw[(r + half * 8) * PSTR +      l16] = s[2 * hv + 0][r];
        Pw[(r + half * 8) * PSTR + 16 + l16] = s[2 * hv + 1][r];
      }
#pragma unroll
      for (int kb = 0; kb < 8; ++kb) {
        const v2f pf = *(const v2f*)(&Pw[l16 * PSTR + kb * 4 + half * 2]);
        const int krow = (hv * 8 + kb) * 4 + half * 2;
        v2f b0, b1, b2, b3;
        b0.x = Vs[krow * LSTR +  0 + l16]; b0.y = Vs[(krow + 1) * LSTR +  0 + l16];
        b1.x = Vs[krow * LSTR + 16 + l16]; b1.y = Vs[(krow + 1) * LSTR + 16 + l16];
        b2.x = Vs[krow * LSTR + 32 + l16]; b2.y = Vs[(krow + 1) * LSTR + 32 + l16];
        b3.x = Vs[krow * LSTR + 48 + l16]; b3.y = Vs[(krow + 1) * LSTR + 48 + l16];
        o0 = wmma_f32(pf, b0, o0);
        o1 = wmma_f32(pf, b1, o1);
        o2 = wmma_f32(pf, b2, o2);
        o3 = wmma_f32(pf, b3, o3);
      }
    }
  }

  // Normalize and write concat-head layout [B,T,D]
#pragma unroll
  for (int r = 0; r < 8; ++r) {
    const float inv = 1.f / lsum[r];
    const int t = q0 + wave * 16 + half * 8 + r;
    const long long base = ((long long)(b * T_ + t)) * D_ + h * DK_;
    AttOut[base +  0 + l16] = o0[r] * inv;
    AttOut[base + 16 + l16] = o1[r] * inv;
    AttOut[base + 32 + l16] = o2[r] * inv;
    AttOut[base + 48 + l16] = o3[r] * inv;
  }
}

// ---------------------------------------------------------------------------
extern "C" void kernel_launch(void* const* d_in, const int* in_sizes, int n_in,
                              void* d_out, int out_size, void* d_ws, size_t ws_size,
                              hipStream_t stream) {
  (void)in_sizes; (void)n_in; (void)out_size; (void)ws_size;
  const float* x  = (const float*)d_in[0];
  const float* wq = (const float*)d_in[1];
  const float* bq = (const float*)d_in[2];
  const float* wk = (const float*)d_in[3];
  const float* bk = (const float*)d_in[4];
  const float* wv = (const float*)d_in[5];
  const float* bv = (const float*)d_in[6];
  const float* wo = (const float*)d_in[7];
  const float* bo = (const float*)d_in[8];
  float* out = (float*)d_out;

  float* Qw  = (float*)d_ws;
  const size_t perBH = (size_t)B_ * H_ * T_ * DK_;   // 8,388,608 floats
  float* Kw  = Qw + perBH;
  float* Vw  = Kw + perBH;
  float* Att = Vw + perBH;

  const dim3 blk(256);

  // QKV projections: z = b*H + h; out [B,H,T,DK]; Q pre-scaled by 1/sqrt(DK)
  const dim3 gqkv(T_ / 128, DK_ / 64, B_ * H_);
  gemm_xwT_kernel<<<gqkv, blk, 0, stream>>>(
      x, wq, bq, Qw, T_, DK_, D_, 0.125f,
      H_, (long long)T_ * D_, H_, (long long)DK_ * D_, DK_, (long long)T_ * DK_);
  gemm_xwT_kernel<<<gqkv, blk, 0, stream>>>(
      x, wk, bk, Kw, T_, DK_, D_, 1.0f,
      H_, (long long)T_ * D_, H_, (long long)DK_ * D_, DK_, (long long)T_ * DK_);
  gemm_xwT_kernel<<<gqkv, blk, 0, stream>>>(
      x, wv, bv, Vw, T_, DK_, D_, 1.0f,
      H_, (long long)T_ * D_, H_, (long long)DK_ * D_, DK_, (long long)T_ * DK_);

  // Causal flash attention -> Att in [B,T,D] concat-head layout
  const dim3 gatt(T_ / 128, B_ * H_);
  flash_attn_kernel<<<gatt, blk, 0, stream>>>(Qw, Kw, Vw, Att);

  // Output projection: out = Att * wo^T + bo
  const dim3 gout((B_ * T_) / 128, D_ / 64, 1);
  gemm_xwT_kernel<<<gout, blk, 0, stream>>>(
      Att, wo, bo, out, B_ * T_, D_, D_, 1.0f,
      1, 0, 1, 0, 0, 0);
}